// S4Kernel_86681029968520
// MI455X (gfx1250) — compile-verified
//
#include <hip/hip_runtime.h>
#include <math.h>

#define S4L 16384
#define S4N 256
#define S4PI 3.14159265358979323846f

typedef __attribute__((ext_vector_type(2))) float v2f;
typedef __attribute__((ext_vector_type(8))) float v8f;

// D = A(16x4 f32) * B(4x16 f32) + C(16x16 f32), full-precision CDNA5 WMMA.
static __device__ __forceinline__ v8f s4_wmma4(v2f a, v2f b, v8f c) {
  return __builtin_amdgcn_wmma_f32_16x16x4_f32(false, a, false, b, (short)0, c,
                                               false, false);
}

// ---------- prep: DFT128 table (W128^{rc} = exp(+2pi i rc/128)), zero pads ----------
__global__ void s4_prep(const float* __restrict__ u,
                        float* __restrict__ dftRe, float* __restrict__ dftIm,
                        float* __restrict__ KrevPad, float* __restrict__ uPad) {
  int i = blockIdx.x * blockDim.x + threadIdx.x;
  if (i < 128 * 128) {
    int r = i >> 7, c = i & 127;
    float ang = (2.0f * S4PI / 128.0f) * (float)((r * c) & 127);
    float s, co;
    __sincosf(ang, &s, &co);
    dftRe[i] = co;
    dftIm[i] = s;
  }
  if (i < S4L) KrevPad[i] = 0.0f;              // lower guard half of reversed K
  if (i < S4L + 256) uPad[i] = (i < S4L) ? u[i] : 0.0f;
}

// ---------- frequency-domain DPLR evaluation: at_roots[k] ----------
__global__ void s4_at_roots(const float* __restrict__ Lre, const float* __restrict__ Lim,
                            const float* __restrict__ Pre, const float* __restrict__ Pim,
                            const float* __restrict__ Bre, const float* __restrict__ Bim,
                            const float* __restrict__ Cre, const float* __restrict__ Cim,
                            const float* __restrict__ log_dt,
                            float* __restrict__ atRe, float* __restrict__ atIm) {
  __shared__ float sv[8][S4N];
  __shared__ float slr[S4N], sli[S4N];
  for (int n = threadIdx.x; n < S4N; n += blockDim.x) {
    float pr = Pre[n], pi = Pim[n], br = Bre[n], bi = Bim[n];
    float cr = Cre[n], ci = Cim[n];
    sv[0][n] = cr * br + ci * bi;  sv[1][n] = cr * bi - ci * br;   // conj(C)*B
    sv[2][n] = cr * pr + ci * pi;  sv[3][n] = cr * pi - ci * pr;   // conj(C)*P
    sv[4][n] = pr * br + pi * bi;  sv[5][n] = pr * bi - pi * br;   // conj(P)*B
    sv[6][n] = pr * pr + pi * pi;  sv[7][n] = 0.0f;                // conj(P)*P (real)
    slr[n] = fminf(Lre[n], -1.0e-4f);
    sli[n] = Lim[n];
  }
  __syncthreads();
  int k = blockIdx.x * blockDim.x + threadIdx.x;
  float step = __expf(log_dt[0]);
  float theta = -2.0f * S4PI * (float)k / (float)S4L;
  float sn, cs;
  __sincosf(theta, &sn, &cs);                  // Omega = cs + i*sn
  float ar = 1.0f - cs, ai = -sn;              // 1 - Omega
  float br = 1.0f + cs, bi = sn;               // 1 + Omega
  float ib = 1.0f / (br * br + bi * bi);
  float t2 = 2.0f / step;
  float gr = t2 * (ar * br + ai * bi) * ib;    // g = (2/step)(1-O)/(1+O)
  float gi = t2 * (ai * br - ar * bi) * ib;
  float c2r = 2.0f * br * ib, c2i = -2.0f * bi * ib;   // c = 2/(1+O)
  float k00r = 0, k00i = 0, k01r = 0, k01i = 0, k10r = 0, k10i = 0, k11r = 0, k11i = 0;
#pragma unroll 4
  for (int n = 0; n < S4N; ++n) {
    float dr = gr - slr[n], di = gi - sli[n];
    float idn = 1.0f / (dr * dr + di * di);
    float rr = dr * idn, ri = -di * idn;       // 1/(g - Lambda)
    k00r += rr * sv[0][n] - ri * sv[1][n];  k00i += rr * sv[1][n] + ri * sv[0][n];
    k01r += rr * sv[2][n] - ri * sv[3][n];  k01i += rr * sv[3][n] + ri * sv[2][n];
    k10r += rr * sv[4][n] - ri * sv[5][n];  k10i += rr * sv[5][n] + ri * sv[4][n];
    k11r += rr * sv[6][n];                  k11i += ri * sv[6][n];
  }
  float denr = 1.0f + k11r, deni = k11i;
  float idn2 = 1.0f / (denr * denr + deni * deni);
  float tr = k01r * k10r - k01i * k10i, ti = k01r * k10i + k01i * k10r;
  float qr = (tr * denr + ti * deni) * idn2;
  float qi = (ti * denr - tr * deni) * idn2;
  float sr = k00r - qr, si = k00i - qi;
  atRe[k] = c2r * sr - c2i * si;
  atIm[k] = c2r * si + c2i * sr;
}

// ---------- IFFT stage 1: G[c][b] = sum_a W128^{ca} at[128a+b], then twiddle W^{cb} ----
// One wave32 per 16x16 output tile. Complex GEMM = 4 f32 WMMA accumulators.
__global__ void s4_fft_stage1(const float* __restrict__ dftRe, const float* __restrict__ dftIm,
                              const float* __restrict__ atRe, const float* __restrict__ atIm,
                              float* __restrict__ Gre, float* __restrict__ Gim) {
  int l = threadIdx.x;
  int mrow = l & 15;                 // A row (M)
  int ncol = l & 15;                 // B col (N)
  int kb = (l >> 4) << 1;            // K pair base: lanes 0-15 -> k{0,1}, 16-31 -> k{2,3}
  int chalf = l >> 4;                // C/D: M = j + 8*chalf
  int c0 = (blockIdx.x >> 3) * 16;
  int b0 = (blockIdx.x & 7) * 16;
  v8f accRR = {}, accII = {}, accRI = {}, accIR = {};
  for (int a0 = 0; a0 < 128; a0 += 4) {
    int ab = (c0 + mrow) * 128 + a0 + kb;
    v2f aR, aI, bR, bI;
    aR[0] = dftRe[ab];  aR[1] = dftRe[ab + 1];
    aI[0] = dftIm[ab];  aI[1] = dftIm[ab + 1];
    int bb = (a0 + kb) * 128 + b0 + ncol;
    bR[0] = atRe[bb];   bR[1] = atRe[bb + 128];
    bI[0] = atIm[bb];   bI[1] = atIm[bb + 128];
    accRR = s4_wmma4(aR, bR, accRR);
    accII = s4_wmma4(aI, bI, accII);
    accRI = s4_wmma4(aR, bI, accRI);
    accIR = s4_wmma4(aI, bR, accIR);
  }
#pragma unroll
  for (int j = 0; j < 8; ++j) {
    int c = c0 + j + 8 * chalf;
    int b = b0 + ncol;
    float gr = accRR[j] - accII[j];
    float gi = accRI[j] + accIR[j];
    float ang = (2.0f * S4PI / (float)S4L) * (float)(c * b);   // twiddle W^{cb}
    float s, co;
    __sincosf(ang, &s, &co);
    Gre[c * 128 + b] = gr * co - gi * s;
    Gim[c * 128 + b] = gr * s + gi * co;
  }
}

// ---------- IFFT stage 2: K[c+128d] = (1/L) Re( sum_b W128^{db} H[c][b] ) ----------
// Real part only -> 2 WMMA accumulators. Writes K reversed into padded buffer:
// KrevPad[2L-1-idx] = K[idx]   (lower half already zeroed by prep).
__global__ void s4_fft_stage2(const float* __restrict__ dftRe, const float* __restrict__ dftIm,
                              const float* __restrict__ Gre, const float* __restrict__ Gim,
                              float* __restrict__ KrevPad) {
  int l = threadIdx.x;
  int mrow = l & 15;
  int ncol = l & 15;
  int kb = (l >> 4) << 1;
  int chalf = l >> 4;
  int d0 = (blockIdx.x >> 3) * 16;
  int c0 = (blockIdx.x & 7) * 16;
  v8f accRR = {}, accII = {};
  for (int b0 = 0; b0 < 128; b0 += 4) {
    int ab = (d0 + mrow) * 128 + b0 + kb;
    v2f aR, aI, bR, bI;
    aR[0] = dftRe[ab];  aR[1] = dftRe[ab + 1];
    aI[0] = dftIm[ab];  aI[1] = dftIm[ab + 1];
    int bb = (c0 + ncol) * 128 + b0 + kb;     // B[k][n] = H[c0+n][b0+k] (transposed read)
    bR[0] = Gre[bb];    bR[1] = Gre[bb + 1];
    bI[0] = Gim[bb];    bI[1] = Gim[bb + 1];
    accRR = s4_wmma4(aR, bR, accRR);
    accII = s4_wmma4(aI, bI, accII);
  }
#pragma unroll
  for (int j = 0; j < 8; ++j) {
    int d = d0 + j + 8 * chalf;
    int c = c0 + ncol;
    int idx = c + 128 * d;
    KrevPad[2 * S4L - 1 - idx] = (accRR[j] - accII[j]) * (1.0f / (float)S4L);
  }
}

// ---------- Linear correlation as zero-waste Toeplitz-block WMMA GEMM ----------
// y[i0+16n+m] = sum_t u[t] * Krev[i - t];  A[m][k]=KrevPad[L+i0+m-t0-k], B[k][n]=uPad[t0+16n+k]
__global__ void s4_corr(const float* __restrict__ uPad, const float* __restrict__ KrevPad,
                        const float* __restrict__ u, const float* __restrict__ Dsc,
                        float* __restrict__ y) {
  __shared__ float red[16][32][8];
  int lane = threadIdx.x & 31;
  int wave = threadIdx.x >> 5;
  int mrow = lane & 15;
  int ncol = lane & 15;
  int kb = (lane >> 4) << 1;
  int i0 = blockIdx.x * 256;
  v8f acc = {};
  // 16 waves stripe the K-dimension (t0); 256 WMMA iterations per wave.
  for (int t0 = wave * 4; t0 < S4L; t0 += 64) {
    int ab = S4L + i0 + mrow - t0 - kb;
    v2f a, b;
    a[0] = KrevPad[ab];
    a[1] = KrevPad[ab - 1];
    int bb = t0 + 16 * ncol + kb;
    b[0] = uPad[bb];
    b[1] = uPad[bb + 1];
    __builtin_prefetch(uPad + bb + 64, 0, 0);   // global_prefetch_b8
    acc = s4_wmma4(a, b, acc);
  }
#pragma unroll
  for (int j = 0; j < 8; ++j) red[wave][lane][j] = acc[j];
  __syncthreads();
  if (threadIdx.x < 256) {
    int l2 = threadIdx.x & 31;
    int j2 = threadIdx.x >> 5;
    float s = 0.0f;
#pragma unroll
    for (int w = 0; w < 16; ++w) s += red[w][l2][j2];
    int m = j2 + 8 * (l2 >> 4);
    int n = l2 & 15;
    int idx = i0 + 16 * n + m;
    y[idx] = s + Dsc[0] * u[idx];
  }
}

extern "C" void kernel_launch(void* const* d_in, const int* in_sizes, int n_in,
                              void* d_out, int out_size, void* d_ws, size_t ws_size,
                              hipStream_t stream) {
  const float* u      = (const float*)d_in[0];
  const float* Lre    = (const float*)d_in[1];
  const float* Lim    = (const float*)d_in[2];
  const float* Pre    = (const float*)d_in[3];
  const float* Pim    = (const float*)d_in[4];
  const float* Bre    = (const float*)d_in[5];
  const float* Bim    = (const float*)d_in[6];
  const float* Cre    = (const float*)d_in[7];
  const float* Cim    = (const float*)d_in[8];
  const float* Dsc    = (const float*)d_in[9];
  const float* log_dt = (const float*)d_in[10];
  float* y = (float*)d_out;

  float* ws      = (float*)d_ws;          // ~590 KB of fp32 scratch, rebuilt every call
  float* atRe    = ws + 0 * S4L;
  float* atIm    = ws + 1 * S4L;
  float* dftRe   = ws + 2 * S4L;          // 128x128 DFT matrix (= 16384 = S4L entries)
  float* dftIm   = ws + 3 * S4L;
  float* Gre     = ws + 4 * S4L;
  float* Gim     = ws + 5 * S4L;
  float* KrevPad = ws + 6 * S4L;          // 2*S4L floats (reversed K, zero-guarded)
  float* uPad    = ws + 8 * S4L;          // S4L + 256 floats

  s4_prep<<<65, 256, 0, stream>>>(u, dftRe, dftIm, KrevPad, uPad);
  s4_at_roots<<<S4L / 256, 256, 0, stream>>>(Lre, Lim, Pre, Pim, Bre, Bim, Cre, Cim,
                                             log_dt, atRe, atIm);
  s4_fft_stage1<<<64, 32, 0, stream>>>(dftRe, dftIm, atRe, atIm, Gre, Gim);
  s4_fft_stage2<<<64, 32, 0, stream>>>(dftRe, dftIm, Gre, Gim, KrevPad);
  s4_corr<<<S4L / 256, 512, 0, stream>>>(uPad, KrevPad, u, Dsc, y);

  (void)in_sizes; (void)n_in; (void)out_size; (void)ws_size;
}